// M2RNN_88304527606209
// MI455X (gfx1250) — compile-verified
//
#include <hip/hip_runtime.h>
#include <hip/hip_bf16.h>
#include <math.h>

typedef float v2f __attribute__((ext_vector_type(2)));
typedef float v8f __attribute__((ext_vector_type(8)));

// Problem constants
#define BB     2
#define SS     1024
#define DIN    1024
#define NH     16
#define KH     64
#define VH     64
#define KSZ    4
#define CONVD  3072      // Q+K+V = 1024*3
#define XPD    4112      // CONVD + N + G
#define GOFF   3088      // CONVD + N
#define GSH    1024
#define DOUT   1024
#define EPSN   1e-6f

// ---------------------------------------------------------------------------
// Generic f32 WMMA GEMM:  C[M x Nout] = A[M x 1024] * W[Nout x 1024]^T
// One 16x16 tile per wave, K chained through 256 v_wmma_f32_16x16x4_f32.
// A-frag: lane(0-15)=row m, K pair {0,1}; lane(16-31)=row m, K pair {2,3}.
// B-frag: lane(0-15)=col n, K pair {0,1}; lane(16-31)=col n, K pair {2,3}.
// global_prefetch_b8 issued ~6 unroll-blocks ahead to keep L2->WGP ahead of
// the chained WMMA stream.
// ---------------------------------------------------------------------------
__global__ __launch_bounds__(256)
void gemm_nt_f32(const float* __restrict__ A, int lda,
                 const float* __restrict__ Wt, int ldw,
                 float* __restrict__ C, int ldc,
                 int tilesM, int tilesN)
{
    int waveId = threadIdx.x >> 5;
    int tile   = blockIdx.x * 8 + waveId;
    int total  = tilesM * tilesN;
    if (tile >= total) return;                       // wave-uniform exit
    int tm = tile / tilesN;
    int tn = tile % tilesN;
    int lane = threadIdx.x & 31;
    int hi   = (lane >> 4) & 1;
    int nl   = lane & 15;

    const float* arow = A  + (size_t)(tm * 16 + nl) * lda + 2 * hi;
    const float* brow = Wt + (size_t)(tn * 16 + nl) * ldw + 2 * hi;

    v8f acc = {0.f, 0.f, 0.f, 0.f, 0.f, 0.f, 0.f, 0.f};
    for (int kko = 0; kko < 32; ++kko) {
        // speculative prefetch ~6 blocks (192 floats) ahead
        __builtin_prefetch(arow + (kko * 8 + 48) * 4, 0, 3);
        __builtin_prefetch(brow + (kko * 8 + 48) * 4, 0, 3);
#pragma unroll
        for (int kki = 0; kki < 8; ++kki) {
            int kk = kko * 8 + kki;
            v2f a = *(const v2f*)(arow + kk * 4);
            v2f b = *(const v2f*)(brow + kk * 4);
            acc = __builtin_amdgcn_wmma_f32_16x16x4_f32(false, a, false, b,
                                                        (short)0, acc, false, false);
        }
    }
    float* crow = C + (size_t)(tm * 16 + 8 * hi) * ldc + tn * 16 + nl;
#pragma unroll
    for (int r = 0; r < 8; ++r) crow[(size_t)r * ldc] = acc[r];
}

// ---------------------------------------------------------------------------
// Depthwise causal conv (KS=4) + SiLU over (B,S,CONVD)
// ---------------------------------------------------------------------------
__global__ __launch_bounds__(256)
void conv_silu_kernel(const float* __restrict__ xp,
                      const float* __restrict__ conv_w,
                      float* __restrict__ xc)
{
    int idx = blockIdx.x * 256 + threadIdx.x;
    if (idx >= BB * SS * CONVD) return;
    int c = idx % CONVD;
    int s = (idx / CONVD) % SS;
    int b = idx / (CONVD * SS);
    float acc = 0.f;
#pragma unroll
    for (int j = 0; j < KSZ; ++j) {
        int sp = s + j - (KSZ - 1);
        if (sp >= 0)
            acc += xp[((size_t)(b * SS + sp)) * XPD + c] * conv_w[c * KSZ + j];
    }
    float sig = 1.f / (1.f + expf(-acc));
    xc[idx] = acc * sig;
}

// ---------------------------------------------------------------------------
// f = exp(-softplus(f_raw + dt_bias) * exp(A_log)),  per (b,s,n)
// ---------------------------------------------------------------------------
__global__ __launch_bounds__(256)
void fgate_kernel(const float* __restrict__ xp,
                  const float* __restrict__ A_log,
                  const float* __restrict__ dt_bias,
                  float* __restrict__ fbuf)
{
    int idx = blockIdx.x * 256 + threadIdx.x;
    if (idx >= BB * SS * NH) return;
    int n = idx % NH;
    int s = (idx / NH) % SS;
    int b = idx / (NH * SS);
    float fr = xp[((size_t)(b * SS + s)) * XPD + CONVD + n] + dt_bias[n];
    float sp = (fr > 20.f) ? fr : log1pf(expf(fr));
    fbuf[idx] = expf(-sp * expf(A_log[n]));
}

// ---------------------------------------------------------------------------
// Sequential scan. One block per (b,n): 512 threads = 16 waves.
// 64x64 f32 state ping-ponged between two LDS buffers (removes the
// read-before-write barrier: 2 barriers/step instead of 3).
// Global inputs for step s+1 are loaded into registers during step s's
// compute phase (register double-buffering), so the per-step critical path
// is: LDS staging -> barrier -> 16 chained f32 WMMAs -> tanh/blend ->
// ds_add reduction -> barrier -> 256B output store.
// ---------------------------------------------------------------------------
__global__ __launch_bounds__(512)
void scan_kernel(const float* __restrict__ xc,
                 const float* __restrict__ fbuf,
                 const float* __restrict__ sw_all,   // (N,64,64)
                 const float* __restrict__ D_res,    // (N,64)
                 float* __restrict__ yscan)          // (B,S,N*64)
{
    int b = blockIdx.x >> 4;
    int n = blockIdx.x & 15;
    int tid = threadIdx.x;
    int w   = tid >> 5;
    int ti  = w >> 2;
    int tj  = w & 3;
    int lane = tid & 31;
    int hi   = (lane >> 4) & 1;
    int nl   = lane & 15;

    __shared__ float Hs[2][64 * 64];
    __shared__ float kb[64], vb[64], qb[64], yp[64];
    __shared__ float fs;

    for (int i = tid; i < 2 * 64 * 64; i += 512) (&Hs[0][0])[i] = 0.f;

    // Preload B-fragments of state_weight column block (constant over s)
    const float* sw = sw_all + (size_t)n * 64 * 64;
    int u = tj * 16 + nl;
    v2f Bf[16];
#pragma unroll
    for (int kk = 0; kk < 16; ++kk) {
        int v0 = kk * 4 + 2 * hi;
        Bf[kk][0] = sw[(size_t)v0 * 64 + u];
        Bf[kk][1] = sw[(size_t)(v0 + 1) * 64 + u];
    }
    float dreg = (tid < 64) ? D_res[n * 64 + tid] : 0.f;
    const int rowA = ti * 16 + nl;      // A-fragment row for this lane

    // Register-prefetch step 0
    float kreg = 0.f, vreg = 0.f, qreg = 0.f, freg = 0.f;
    {
        size_t base0 = (size_t)b * SS;
        if (tid < 64) {
            kreg = xc[base0 * CONVD + 1024 + n * 64 + tid];
            vreg = xc[base0 * CONVD + 2048 + n * 64 + tid];
        } else if (tid < 128) {
            qreg = xc[base0 * CONVD + n * 64 + (tid - 64)];
        } else if (tid == 128) {
            freg = fbuf[base0 * NH + n];
        }
    }
    __syncthreads();

    for (int s = 0; s < SS; ++s) {
        int p = s & 1;
        float vcur = 0.f;
        // ---- stage current step from registers into LDS ----
        if (tid < 64) {
            kb[tid] = kreg;
            vcur    = vreg;
            vb[tid] = vreg;
            yp[tid] = 0.f;
        } else if (tid < 128) {
            qb[tid - 64] = qreg;
        } else if (tid == 128) {
            fs = freg;
        }
        // ---- issue global loads for step s+1 (consumed next iteration) ----
        if (s + 1 < SS) {
            size_t bn = (size_t)b * SS + (s + 1);
            if (tid < 64) {
                kreg = xc[bn * CONVD + 1024 + n * 64 + tid];
                vreg = xc[bn * CONVD + 2048 + n * 64 + tid];
            } else if (tid < 128) {
                qreg = xc[bn * CONVD + n * 64 + (tid - 64)];
            } else if (tid == 128) {
                freg = fbuf[bn * NH + n];
            }
        }
        __syncthreads();   // barrier A: staging visible, prior-step writes safe

        // acc = k (x) v  for this tile
        float vv = vb[tj * 16 + nl];
        v8f acc;
#pragma unroll
        for (int r = 0; r < 8; ++r)
            acc[r] = kb[ti * 16 + r + 8 * hi] * vv;

        // acc += H_row_block @ W_col_block  (K=64 -> 16 chained f32 WMMAs)
#pragma unroll
        for (int kk = 0; kk < 16; ++kk) {
            v2f a = *(const v2f*)(&Hs[p][(size_t)rowA * 64 + kk * 4 + 2 * hi]);
            acc = __builtin_amdgcn_wmma_f32_16x16x4_f32(false, a, false, Bf[kk],
                                                        (short)0, acc, false, false);
        }

        float f = fs;
        float ypart = 0.f;
#pragma unroll
        for (int r = 0; r < 8; ++r) {
            int m = ti * 16 + r + 8 * hi;
            float hold = Hs[p][(size_t)m * 64 + tj * 16 + nl];
            float hn   = tanhf(acc[r]);
            float hv   = f * hold + (1.f - f) * hn;
            Hs[p ^ 1][(size_t)m * 64 + tj * 16 + nl] = hv;   // write other buffer
            ypart += qb[m] * hv;
        }
        atomicAdd(&yp[tj * 16 + nl], ypart);   // ds_add_f32 reduction over ti
        __syncthreads();   // barrier B: new state + y reduction complete

        if (tid < 64)
            yscan[((size_t)b * SS + s) * (NH * VH) + n * 64 + tid] =
                yp[tid] + vcur * dreg;
    }
}

// ---------------------------------------------------------------------------
// y = yscan * silu(g); RMS-norm over 1024 dims; * g_norm_w
// One block per (b*S+s) row.
// ---------------------------------------------------------------------------
__global__ __launch_bounds__(256)
void norm_kernel(const float* __restrict__ yscan,
                 const float* __restrict__ xp,
                 const float* __restrict__ gnw,
                 float* __restrict__ yn)
{
    int row = blockIdx.x;
    int tid = threadIdx.x;
    __shared__ float sred[256];
    float vals[4];
    float sum = 0.f;
#pragma unroll
    for (int e = 0; e < 4; ++e) {
        int idx = e * 256 + tid;
        float yv = yscan[(size_t)row * GSH + idx];
        float gv = xp[(size_t)row * XPD + GOFF + idx];
        float gs = gv / (1.f + expf(-gv));   // silu(g)
        float t  = yv * gs;
        vals[e] = t;
        sum += t * t;
    }
    sred[tid] = sum;
    __syncthreads();
    for (int off = 128; off > 0; off >>= 1) {
        if (tid < off) sred[tid] += sred[tid + off];
        __syncthreads();
    }
    float rs = rsqrtf(sred[0] * (1.f / (float)GSH) + EPSN);
#pragma unroll
    for (int e = 0; e < 4; ++e) {
        int idx = e * 256 + tid;
        yn[(size_t)row * GSH + idx] = vals[e] * rs * gnw[idx];
    }
}

// ---------------------------------------------------------------------------
extern "C" void kernel_launch(void* const* d_in, const int* in_sizes, int n_in,
                              void* d_out, int out_size, void* d_ws, size_t ws_size,
                              hipStream_t stream)
{
    const float* x       = (const float*)d_in[0];
    const float* W_in    = (const float*)d_in[1];
    const float* conv_w  = (const float*)d_in[2];
    const float* A_log   = (const float*)d_in[3];
    const float* dt_bias = (const float*)d_in[4];
    const float* sw      = (const float*)d_in[5];
    const float* D_res   = (const float*)d_in[6];
    const float* g_norm  = (const float*)d_in[7];
    const float* W_out   = (const float*)d_in[8];
    float* out = (float*)d_out;

    // Workspace layout (floats)
    float* xp    = (float*)d_ws;                       // B*S*4112
    float* xc    = xp    + (size_t)BB * SS * XPD;      // B*S*3072
    float* fbuf  = xc    + (size_t)BB * SS * CONVD;    // B*S*16
    float* yscan = fbuf  + (size_t)BB * SS * NH;       // B*S*1024
    float* yn    = yscan + (size_t)BB * SS * GSH;      // B*S*1024

    // 1) xp = x @ W_in^T   (2048x1024 * 4112x1024^T): 128 x 257 tiles
    gemm_nt_f32<<<(128 * 257 + 7) / 8, 256, 0, stream>>>(
        x, DIN, W_in, DIN, xp, XPD, 128, 257);

    // 2) causal conv + SiLU
    conv_silu_kernel<<<(BB * SS * CONVD + 255) / 256, 256, 0, stream>>>(
        xp, conv_w, xc);

    // 3) decay gate f
    fgate_kernel<<<(BB * SS * NH + 255) / 256, 256, 0, stream>>>(
        xp, A_log, dt_bias, fbuf);

    // 4) sequential recurrence (32 independent (b,n) recurrences)
    scan_kernel<<<BB * NH, 512, 0, stream>>>(xc, fbuf, sw, D_res, yscan);

    // 5) gate + RMS norm
    norm_kernel<<<BB * SS, 256, 0, stream>>>(yscan, xp, g_norm, yn);

    // 6) out = yn @ W_out^T  (2048x1024 * 1024x1024^T): 128 x 64 tiles
    gemm_nt_f32<<<(128 * 64 + 7) / 8, 256, 0, stream>>>(
        yn, GSH, W_out, GSH, out, DOUT, 128, 64);
}